// LModel21_13537736917365
// MI455X (gfx1250) — compile-verified
//
#include <hip/hip_runtime.h>
#include <math.h>

typedef __attribute__((ext_vector_type(2))) float v2f;
typedef __attribute__((ext_vector_type(8))) float v8f;

// ---------------------------------------------------------------------------
// Kernel 1: project all 2*N_EDGES endpoints through the 5 (3x4) matrices with
// V_WMMA_F32_16X16X4_F32.  A = 16x4 (rows 0..2 intrinsic|0, rows 3..14 the 4
// source transforms, row 15 zero).  B = 4x16 homogeneous points.  D rows
// 3v..3v+2 give (u_num, v_num, w) for view v of each of the 16 point columns.
// ISA layouts (7.12.2): A: lanes0-15 -> M rows, VGPR0 = K0|K2, VGPR1 = K1|K3.
// B mirrors with N across lanes; D: lane n (n=lane&15), VGPR r -> M = r+8*(lane>>4).
// ---------------------------------------------------------------------------
__device__ __forceinline__ float proj_mat_elem(const float* __restrict__ intr,
                                               const float* __restrict__ trans,
                                               int r, int c, int num_src) {
    if (r < 3) return (c < 3) ? intr[r * 3 + c] : 0.0f;
    int rr = r - 3;
    if (rr < num_src * 3) return trans[rr * 4 + c];
    return 0.0f;
}

__global__ void proj_wmma_kernel(const float* __restrict__ start,
                                 const float* __restrict__ endp,
                                 const float* __restrict__ intr,
                                 const float* __restrict__ trans,
                                 float* __restrict__ proj_ws,
                                 int n_points, int num_src) {
    const int lane = threadIdx.x & 31;
    const int base = blockIdx.x * 16;
    const int row  = lane & 15;     // A matrix row / B,D column
    const int half = lane >> 4;
    const int kb   = half * 2;      // K base for this half-wave

    v2f a;
    a[0] = proj_mat_elem(intr, trans, row, kb,     num_src);
    a[1] = proj_mat_elem(intr, trans, row, kb + 1, num_src);

    // B column = point (base+row) as [x,y,z,1]
    float pc0 = 0.f, pc1 = 0.f, pc2 = 0.f, pc3 = 0.f;
    int p = base + row;
    if (p < n_points) {
        int e = p >> 1;
        const float* src = (p & 1) ? endp : start;
        pc0 = src[e * 3 + 0];
        pc1 = src[e * 3 + 1];
        pc2 = src[e * 3 + 2];
        pc3 = 1.0f;
    }
    v2f b;
    b[0] = (kb == 0) ? pc0 : pc2;
    b[1] = (kb == 0) ? pc1 : pc3;

    v8f c = {0.f, 0.f, 0.f, 0.f, 0.f, 0.f, 0.f, 0.f};
    v8f d = __builtin_amdgcn_wmma_f32_16x16x4_f32(
        /*neg_a=*/false, a, /*neg_b=*/false, b,
        /*c_mod=*/(short)0, c, /*reuse_a=*/false, /*reuse_b=*/false);

    int pp = base + (lane & 15);
    if (pp < n_points) {
#pragma unroll
        for (int r8 = 0; r8 < 8; ++r8) {
            int m = r8 + 8 * half;                 // matrix output row
            if (m < 15) proj_ws[pp * 16 + m] = d[r8];
        }
    }
}

// ---------------------------------------------------------------------------
// Kernel 2: per (view, edge) template parameters:
//   ep0, cur_dir, edge_up  ->  6 floats
// ---------------------------------------------------------------------------
__global__ void edge_params_kernel(const float* __restrict__ proj_ws,
                                   float* __restrict__ params,
                                   int n_edges, int n_views, float up_scale) {
    int idx = blockIdx.x * blockDim.x + threadIdx.x;
    if (idx >= n_edges * n_views) return;
    int v = idx / n_edges;
    int e = idx - v * n_edges;
    const float* P0 = proj_ws + (size_t)(2 * e) * 16 + v * 3;
    const float* P1 = proj_ws + (size_t)(2 * e + 1) * 16 + v * 3;
    float e0x = P0[0] / (P0[2] + 1e-6f);
    float e0y = P0[1] / (P0[2] + 1e-6f);
    float e1x = P1[0] / (P1[2] + 1e-6f);
    float e1y = P1[1] / (P1[2] + 1e-6f);
    float dx = e1x - e0x, dy = e1y - e0y;
    float nrm = sqrtf(dx * dx + dy * dy);      // |cross((dx,dy,0),(0,0,1))|
    float s = up_scale / (nrm + 1e-6f);
    float* o = params + (size_t)idx * 6;
    o[0] = e0x; o[1] = e0y;
    o[2] = dx;  o[3] = dy;
    o[4] = dy * s;  o[5] = -dx * s;            // edge_up = (dy,-dx)/|.| * scale
}

// ---------------------------------------------------------------------------
// Kernel 3: per-sample-point main pass
// ---------------------------------------------------------------------------
__device__ __forceinline__ float bilin(const float* __restrict__ img,
                                       float u, float v, int W, int H) {
    float x = u * (float)W - 0.5f;
    float y = v * (float)H - 0.5f;
    float x0 = floorf(x), y0 = floorf(y);
    float wx = x - x0, wy = y - y0;
    int x0i = min(max((int)x0, 0), W - 1);
    int x1i = min(x0i + 1, W - 1);
    int y0i = min(max((int)y0, 0), H - 1);
    int y1i = min(y0i + 1, H - 1);
    float v00 = img[(size_t)y0i * W + x0i];
    float v01 = img[(size_t)y0i * W + x1i];
    float v10 = img[(size_t)y1i * W + x0i];
    float v11 = img[(size_t)y1i * W + x1i];
    return (v00 * (1.f - wx) + v01 * wx) * (1.f - wy)
         + (v10 * (1.f - wx) + v11 * wx) * wy;
}

__global__ void sample_kernel(const float* __restrict__ imgs,
                              const float* __restrict__ params,
                              const float* __restrict__ coords_x,
                              const float* __restrict__ coords_y,
                              const int* __restrict__ edge_idx,
                              float* __restrict__ out_p2d1,
                              float* __restrict__ out_p2d2,
                              float* __restrict__ sums,
                              float* __restrict__ black,
                              unsigned* __restrict__ masks,
                              int P, int n_edges, int num_src, int W, int H) {
    int i = blockIdx.x * blockDim.x + threadIdx.x;
    bool act = i < P;
    int e = act ? edge_idx[i] : -1;
    float cx = act ? coords_x[i] : 0.f;
    float cy = act ? coords_y[i] : 0.f;

    float s1 = 0.f, blk = 0.f;
    float d2[4] = {0.f, 0.f, 0.f, 0.f};
    unsigned mok[4] = {1u, 1u, 1u, 1u};
    bool vm1 = false;

    if (act) {
        // view 0 (intrinsic)
        const float* pr = params + (size_t)e * 6;
        float px = pr[2] * cx + pr[4] * cy + pr[0];
        float py = pr[3] * cx + pr[5] * cy + pr[1];
        vm1 = (px > 0.f) & (px < 1.f) & (py > 0.f) & (py < 1.f);
        px = fminf(fmaxf(px, 0.f), 0.999999f);
        py = fminf(fmaxf(py, 0.f), 0.999999f);
        out_p2d1[(size_t)i * 2 + 0] = px;
        out_p2d1[(size_t)i * 2 + 1] = py;
        s1 = bilin(imgs, px, py, W, H);
        blk = (s1 < 0.01f) ? 1.f : 0.f;

        for (int s = 0; s < num_src; ++s) {
            const float* qr = params + ((size_t)(s + 1) * n_edges + e) * 6;
            float qx = qr[2] * cx + qr[4] * cy + qr[0];
            float qy = qr[3] * cx + qr[5] * cy + qr[1];
            bool vm2 = (qx > 0.f) & (qx < 1.f) & (qy > 0.f) & (qy < 1.f);
            qx = fminf(fmaxf(qx, 0.f), 0.999999f);
            qy = fminf(fmaxf(qy, 0.f), 0.999999f);
            size_t o = ((size_t)s * P + i) * 2;
            out_p2d2[o + 0] = qx;
            out_p2d2[o + 1] = qy;
            float s2 = bilin(imgs + (size_t)(s + 1) * W * H, qx, qy, W, H);
            float df = s2 - s1;
            d2[s] = df * df;
            mok[s] = (vm1 && vm2) ? 1u : 0u;
        }
    }

    // ---- wave-segmented reduction (edges own >=40 contiguous points) ----
    int e0 = __shfl(e, 0, 32);
    bool uni = __all(act && (e == e0));
    int lane = threadIdx.x & 31;
    if (uni) {
        float acc0 = d2[0], acc1 = d2[1], acc2 = d2[2], acc3 = d2[3], accb = blk;
#pragma unroll
        for (int off = 16; off > 0; off >>= 1) {
            acc0 += __shfl_xor(acc0, off, 32);
            acc1 += __shfl_xor(acc1, off, 32);
            acc2 += __shfl_xor(acc2, off, 32);
            acc3 += __shfl_xor(acc3, off, 32);
            accb += __shfl_xor(accb, off, 32);
        }
        unsigned mall[4];
        for (int s = 0; s < num_src; ++s) mall[s] = __all((int)mok[s]) ? 1u : 0u;
        if (lane == 0) {
            float accs[4] = {acc0, acc1, acc2, acc3};
            for (int s = 0; s < num_src; ++s)
                atomicAdd(&sums[(size_t)s * n_edges + e0], accs[s]);
            atomicAdd(&black[e0], accb);
            for (int s = 0; s < num_src; ++s)
                if (!mall[s]) atomicAnd(&masks[(size_t)s * n_edges + e0], 0u);
        }
    } else if (act) {
        for (int s = 0; s < num_src; ++s)
            atomicAdd(&sums[(size_t)s * n_edges + e], d2[s]);
        atomicAdd(&black[e], blk);
        for (int s = 0; s < num_src; ++s)
            if (!mok[s]) atomicAnd(&masks[(size_t)s * n_edges + e], 0u);
    }
}

// ---------------------------------------------------------------------------
// Kernel 0: reset accumulators (harness poisons d_ws)
// ---------------------------------------------------------------------------
__global__ void init_ws_kernel(float* __restrict__ sums, float* __restrict__ black,
                               unsigned* __restrict__ masks, int n_edges, int num_src) {
    int i = blockIdx.x * blockDim.x + threadIdx.x;
    int tot = num_src * n_edges;
    if (i < tot) { sums[i] = 0.f; masks[i] = 1u; }
    if (i < n_edges) black[i] = 0.f;
}

// ---------------------------------------------------------------------------
// Kernel 4: finalize scalar outputs
// ---------------------------------------------------------------------------
__global__ void finalize_kernel(const float* __restrict__ sums,
                                const float* __restrict__ black,
                                const unsigned* __restrict__ masks,
                                const int* __restrict__ num_per_edge,
                                float* __restrict__ out,
                                int n_edges, int num_src) {
    int e = blockIdx.x * blockDim.x + threadIdx.x;
    if (e >= n_edges) return;
    float cnt = (float)num_per_edge[e];
    for (int s = 0; s < num_src; ++s)
        out[(size_t)s * n_edges + e] = sums[(size_t)s * n_edges + e] / cnt;
    float* om = out + (size_t)num_src * n_edges;
    for (int s = 0; s < num_src; ++s)
        om[(size_t)s * n_edges + e] = masks[(size_t)s * n_edges + e] ? 1.f : 0.f;
    out[(size_t)2 * num_src * n_edges + e] = (black[e] / cnt > 0.5f) ? 1.f : 0.f;
}

// ---------------------------------------------------------------------------
extern "C" void kernel_launch(void* const* d_in, const int* in_sizes, int n_in,
                              void* d_out, int out_size, void* d_ws, size_t ws_size,
                              hipStream_t stream) {
    (void)n_in; (void)out_size; (void)ws_size;
    const float* start = (const float*)d_in[0];
    const float* endp  = (const float*)d_in[1];
    const float* imgs  = (const float*)d_in[2];
    const float* trans = (const float*)d_in[3];
    const float* intr  = (const float*)d_in[4];
    const float* cxs   = (const float*)d_in[5];
    const float* cys   = (const float*)d_in[6];
    const int*   eidx  = (const int*)d_in[7];
    const int*   npe   = (const int*)d_in[8];

    const int n_edges  = in_sizes[0] / 3;           // 6000
    const int num_src  = in_sizes[3] / 12;          // 4
    const int n_views  = num_src + 1;               // 5
    const int n_points = 2 * n_edges;               // 12000
    const int P        = in_sizes[5];               // total sample points
    const int imgHW    = in_sizes[2] / n_views;     // H*W
    const int W        = (int)(sqrt((double)imgHW) + 0.5);
    const int H        = imgHW / W;

    // workspace layout
    float*    ws      = (float*)d_ws;
    float*    proj_ws = ws;                                    // n_points*16
    float*    params  = proj_ws + (size_t)n_points * 16;       // n_views*n_edges*6
    float*    sums    = params + (size_t)n_views * n_edges * 6;// num_src*n_edges
    float*    black   = sums + (size_t)num_src * n_edges;      // n_edges
    unsigned* masks   = (unsigned*)(black + n_edges);          // num_src*n_edges

    float* out      = (float*)d_out;
    float* out_p2d1 = out + (size_t)2 * num_src * n_edges + n_edges;
    float* out_p2d2 = out_p2d1 + (size_t)2 * P;

    {
        int tot = num_src * n_edges;
        init_ws_kernel<<<(tot + 255) / 256, 256, 0, stream>>>(sums, black, masks,
                                                              n_edges, num_src);
    }
    proj_wmma_kernel<<<(n_points + 15) / 16, 32, 0, stream>>>(
        start, endp, intr, trans, proj_ws, n_points, num_src);
    edge_params_kernel<<<(n_views * n_edges + 255) / 256, 256, 0, stream>>>(
        proj_ws, params, n_edges, n_views, 10.0f / (float)W);
    sample_kernel<<<(P + 255) / 256, 256, 0, stream>>>(
        imgs, params, cxs, cys, eidx, out_p2d1, out_p2d2,
        sums, black, masks, P, n_edges, num_src, W, H);
    finalize_kernel<<<(n_edges + 255) / 256, 256, 0, stream>>>(
        sums, black, masks, npe, out, n_edges, num_src);
}